// PredictionsPostProcessing_38654705664162
// MI455X (gfx1250) — compile-verified
//
#include <hip/hip_runtime.h>
#include <stdint.h>

#define BATCH    32
#define NANCH    25200
#define ROWLEN   85
#define NCLS     80
#define TOPK     1000
#define SORTN    2048
#define NTHREADS 1024
#define CHUNK    2048           // elements per async-staged chunk (2 per thread)
#define PER      (CHUNK / NTHREADS)
#define NCHUNK   ((NANCH + CHUNK - 1) / CHUNK)
#define SCORE_T  0.25f

// ---- CDNA5 async global->LDS path (probe-verified present on this toolchain) ----
#if defined(__has_builtin)
#if __has_builtin(__builtin_amdgcn_global_load_async_to_lds_b32)
#define USE_ASYNC 1
typedef __attribute__((address_space(1))) int gas_int;
typedef __attribute__((address_space(3))) int las_int;
__device__ __forceinline__ void async_ld_f32(const float* g, float* l) {
    __builtin_amdgcn_global_load_async_to_lds_b32((gas_int*)g, (las_int*)l, 0, 0);
}
#endif
#endif

__device__ __forceinline__ void wait_async_le0() {
#if defined(__has_builtin) && __has_builtin(__builtin_amdgcn_s_wait_asynccnt)
    __builtin_amdgcn_s_wait_asynccnt(0);
#else
    asm volatile("s_wait_asynccnt 0x0" ::: "memory");
#endif
}
__device__ __forceinline__ void wait_async_le2() {
#if defined(__has_builtin) && __has_builtin(__builtin_amdgcn_s_wait_asynccnt)
    __builtin_amdgcn_s_wait_asynccnt(PER);
#else
    asm volatile("s_wait_asynccnt 0x2" ::: "memory");
#endif
}
__device__ __forceinline__ void wait_ds0() {
    asm volatile("s_wait_dscnt 0x0" ::: "memory");
}

__global__ __launch_bounds__(NTHREADS)
void topk_postproc_kernel(const float* __restrict__ pred,
                          float* __restrict__ out_scores,
                          float* __restrict__ out_boxes) {
    __shared__ unsigned hist[256];
    __shared__ float stage[2 * CHUNK];          // double-buffered async staging
    __shared__ unsigned long long sortbuf[SORTN];
    __shared__ unsigned s_bsel, s_cnt;

    const int tid = threadIdx.x;
    const int b   = blockIdx.x;
    const float* scores = pred + (size_t)b * NANCH * ROWLEN + 4;

    if (tid == 0) s_cnt = 0u;
    if (tid < 256) hist[tid] = 0u;
    for (int j = tid; j < SORTN; j += NTHREADS) sortbuf[j] = 0ull;  // sentinel = smallest key
    __syncthreads();

    // ---- Pass A: single 8-bit MSB histogram over all scores (async double-buffered) ----
#ifdef USE_ASYNC
    {
        // issue chunk 0 into buffer half 0 (clamped addresses keep ASYNCcnt uniform per wave)
        #pragma unroll
        for (int u = 0; u < PER; ++u) {
            int i = tid + u * NTHREADS;
            int ii = (i < NANCH) ? i : (NANCH - 1);
            async_ld_f32(scores + (size_t)ii * ROWLEN, &stage[tid + u * NTHREADS]);
        }
        for (int c = 0; c < NCHUNK; ++c) {
            const int cur = c & 1;
            if (c + 1 < NCHUNK) {
                const int nxt = (c + 1) & 1;
                wait_ds0();   // drain our reads of the buffer we are about to overwrite
                #pragma unroll
                for (int u = 0; u < PER; ++u) {
                    int i = (c + 1) * CHUNK + tid + u * NTHREADS;
                    int ii = (i < NANCH) ? i : (NANCH - 1);
                    async_ld_f32(scores + (size_t)ii * ROWLEN,
                                 &stage[nxt * CHUNK + tid + u * NTHREADS]);
                }
                wait_async_le2();   // in-order completion => chunk c fully in LDS
            } else {
                wait_async_le0();
            }
            #pragma unroll
            for (int u = 0; u < PER; ++u) {
                int i = c * CHUNK + tid + u * NTHREADS;
                if (i < NANCH) {
                    unsigned key =
                        __float_as_uint(((volatile float*)stage)[cur * CHUNK + tid + u * NTHREADS]);
                    atomicAdd(&hist[key >> 24], 1u);
                }
            }
        }
    }
#else
    for (int i = tid; i < NANCH; i += NTHREADS) {
        unsigned key = __float_as_uint(scores[(size_t)i * ROWLEN]);
        atomicAdd(&hist[key >> 24], 1u);
    }
#endif
    __syncthreads();

    // ---- cut bucket: count(topbyte > bsel) < TOPK <= count(topbyte >= bsel) ----
    if (tid == 0) {
        unsigned cum = 0;
        int bsel = 0;
        for (int bb = 255; bb >= 0; --bb) {
            cum += hist[bb];
            if (cum >= TOPK) { bsel = bb; break; }
        }
        s_bsel = (unsigned)bsel;
    }
    __syncthreads();
    const unsigned bsel = s_bsel;

    // ---- Pass B: collect candidate superset (topbyte >= bsel), ~1.1K entries (L2-hot) ----
    for (int i = tid; i < NANCH; i += NTHREADS) {
        unsigned key = __float_as_uint(scores[(size_t)i * ROWLEN]);
        if ((key >> 24) >= bsel) {
            unsigned slot = atomicAdd(&s_cnt, 1u);
            if (slot < SORTN)   // superset >= TOPK guaranteed; overflow needs ~2K-way bucket tie
                sortbuf[slot] = ((unsigned long long)key << 32) | (unsigned)(~(unsigned)i);
        }
    }
    __syncthreads();

    // ---- bitonic sort ascending on (key | ~index): exact top-k, stable ties ----
    for (int kk = 2; kk <= SORTN; kk <<= 1) {
        for (int j = kk >> 1; j > 0; j >>= 1) {
            for (int i = tid; i < SORTN; i += NTHREADS) {
                int ixj = i ^ j;
                if (ixj > i) {
                    unsigned long long a = sortbuf[i];
                    unsigned long long c = sortbuf[ixj];
                    bool asc = ((i & kk) == 0);
                    if ((asc && a > c) || (!asc && a < c)) {
                        sortbuf[i] = c; sortbuf[ixj] = a;
                    }
                }
            }
            __syncthreads();
        }
    }

    // ---- gather + threshold outputs: one wave per selected row (coalesced 80-wide) ----
    const int wid  = tid >> 5;
    const int lane = tid & 31;
    for (int r = wid; r < TOPK; r += NTHREADS / 32) {
        unsigned long long e = sortbuf[SORTN - 1 - r];   // rank r largest
        unsigned keyb = (unsigned)(e >> 32);
        unsigned idx  = ~((unsigned)e);
        float v = __uint_as_float(keyb);
        float s = (v > SCORE_T) ? v : 0.0f;
        const float* row = pred + ((size_t)b * NANCH + idx) * ROWLEN;
        float* so = out_scores + ((size_t)b * TOPK + r) * NCLS;
        for (int c = lane; c < NCLS; c += 32) {
            float m = row[5 + c] * s;
            so[c] = (m > SCORE_T) ? m : 0.0f;
        }
        if (lane < 4) {
            float a = row[lane & 1];
            float h = row[(lane & 1) + 2] * 0.5f;
            out_boxes[((size_t)b * TOPK + r) * 4 + lane] = (lane < 2) ? (a - h) : (a + h);
        }
    }
}

extern "C" void kernel_launch(void* const* d_in, const int* in_sizes, int n_in,
                              void* d_out, int out_size, void* d_ws, size_t ws_size,
                              hipStream_t stream) {
    (void)in_sizes; (void)n_in; (void)out_size; (void)d_ws; (void)ws_size;
    const float* pred = (const float*)d_in[0];
    float* out_scores = (float*)d_out;
    float* out_boxes  = (float*)d_out + (size_t)BATCH * TOPK * NCLS;
    topk_postproc_kernel<<<dim3(BATCH), dim3(NTHREADS), 0, stream>>>(pred, out_scores, out_boxes);
}